// DeformConv_88819923681843
// MI455X (gfx1250) — compile-verified
//
#include <hip/hip_runtime.h>
#include <math.h>

typedef float v2f __attribute__((ext_vector_type(2)));
typedef float v8f __attribute__((ext_vector_type(8)));

#define Hh 128
#define Ww 128
#define HW 16384            // H*W
#define Cc 64
#define Oo 64
#define KK 576              // C*9
#define NSTEP 144           // KK/4 (K per WMMA = 4)
#define MT 32               // pixels (M rows) per block
#define VAL_STRIDE 580      // 576 + 4 pad dwords -> conflict-free b64 reads
#define VAL_FLOATS (MT * VAL_STRIDE)     // 18560 floats
#define WFRAG_COUNT (4 * NSTEP * 32)     // 18432 v2f B-fragments (N=64 -> 4 subtiles)

// ---------------------------------------------------------------------------
// Kernel 1: offset (18ch) + mask (9ch, sigmoid) 3x3 convs, direct form.
// ---------------------------------------------------------------------------
__global__ __launch_bounds__(256)
void dcn_offset_mask_kernel(const float* __restrict__ x,
                            const float* __restrict__ ow, const float* __restrict__ ob,
                            const float* __restrict__ mw, const float* __restrict__ mb,
                            float* __restrict__ offo, float* __restrict__ masko) {
    int pix = blockIdx.x * 256 + threadIdx.x;   // < 32768
    int b  = pix >> 14;
    int hw = pix & (HW - 1);
    int y  = hw >> 7;
    int xx = hw & (Ww - 1);

    float acc[27];
#pragma unroll
    for (int j = 0; j < 18; ++j) acc[j] = ob[j];
#pragma unroll
    for (int j = 0; j < 9; ++j)  acc[18 + j] = mb[j];

    for (int c = 0; c < Cc; ++c) {
        const float* xc = x + (b * Cc + c) * HW;
#pragma unroll
        for (int t = 0; t < 9; ++t) {
            int ty = t / 3, tx = t % 3;
            int yy = y + ty - 1, xw = xx + tx - 1;
            float v = (yy >= 0 && yy < Hh && xw >= 0 && xw < Ww) ? xc[yy * Ww + xw] : 0.f;
#pragma unroll
            for (int j = 0; j < 18; ++j) acc[j]      += v * ow[(j * Cc + c) * 9 + t];
#pragma unroll
            for (int j = 0; j < 9; ++j)  acc[18 + j] += v * mw[(j * Cc + c) * 9 + t];
        }
    }
#pragma unroll
    for (int j = 0; j < 18; ++j) offo[(b * 18 + j) * HW + hw] = acc[j];
#pragma unroll
    for (int j = 0; j < 9; ++j)
        masko[(b * 9 + j) * HW + hw] = 1.f / (1.f + expf(-acc[18 + j]));
}

// ---------------------------------------------------------------------------
// Kernel 2: async weight staging + modulated bilinear sampling into LDS +
//           f32 WMMA GEMM.
//   out_pre[pixel][o] = sum_{c,k} val[pixel][c*9+k] * conv_w[o][c*9+k]
// Block: 256 threads = 8 waves = 2 (M subtiles) x 4 (N subtiles), MT=32 pixels.
// LDS: 18560 f32 val tile + 18432 v2f B-fragments = 221,696 B (one WG/WGP).
// ---------------------------------------------------------------------------
__global__ __launch_bounds__(256)
void dcn_wmma_kernel(const float* __restrict__ x,
                     const float* __restrict__ cw,
                     const float* __restrict__ offb,
                     const float* __restrict__ maskb,
                     float* __restrict__ pre) {
    extern __shared__ float smem[];
    float* valLds = smem;                          // [MT][VAL_STRIDE]
    v2f*   wLds   = (v2f*)(smem + VAL_FLOATS);     // [4][NSTEP][32] B-fragments

    const int tid = threadIdx.x;
    const int gm0 = blockIdx.x * MT;               // first pixel of block (32 | HW -> b uniform)
    const int b   = gm0 >> 14;

    // ---- stage conv_w into LDS pre-swizzled to B-fragment lane layout,
    //      using async global->LDS copies (ASYNCcnt); overlaps with val phase ----
    for (int idx = tid; idx < WFRAG_COUNT; idx += 256) {
        int lane = idx & 31;
        int rest = idx >> 5;
        int step = rest % NSTEP;
        int nw   = rest / NSTEP;
        int n    = nw * 16 + (lane & 15);
        int k0   = 4 * step + 2 * (lane >> 4);
        unsigned lds_off = (unsigned)(VAL_FLOATS * sizeof(float)) + (unsigned)idx * 8u;
        unsigned long long gaddr = (unsigned long long)(const void*)(cw + n * KK + k0);
        asm volatile("global_load_async_to_lds_b64 %0, %1, off"
                     :: "v"(lds_off), "v"(gaddr) : "memory");
    }

    // ---- compute val tile: 32 pixels x 9 taps x 64 channels ----
    for (int p = tid; p < MT * 9; p += 256) {
        int m = p / 9, tap = p % 9;
        int gm = gm0 + m;
        int hw = gm & (HW - 1);
        int yy = hw >> 7, xx = hw & (Ww - 1);
        int ty = tap / 3, tx = tap % 3;

        float dy = offb[(b * 18 + 2 * tap) * HW + hw];
        float dx = offb[(b * 18 + 2 * tap + 1) * HW + hw];
        float ms = maskb[(b * 9 + tap) * HW + hw];

        float py = dy + (float)yy + (float)(ty - 1);
        float px = dx + (float)xx + (float)(tx - 1);
        float y0f = floorf(py), x0f = floorf(px);
        float wy1 = py - y0f,   wx1 = px - x0f;
        float wy0 = 1.f - wy1,  wx0 = 1.f - wx1;
        int iy0 = (int)y0f, ix0 = (int)x0f;
        int iy1 = iy0 + 1,  ix1 = ix0 + 1;
        bool vy0 = (iy0 >= 0) && (iy0 < Hh);
        bool vy1 = (iy1 >= 0) && (iy1 < Hh);
        bool vx0 = (ix0 >= 0) && (ix0 < Ww);
        bool vx1 = (ix1 >= 0) && (ix1 < Ww);
        int cy0 = min(max(iy0, 0), Hh - 1), cy1 = min(max(iy1, 0), Hh - 1);
        int cx0 = min(max(ix0, 0), Ww - 1), cx1 = min(max(ix1, 0), Ww - 1);
        float w00 = (vy0 && vx0) ? wy0 * wx0 * ms : 0.f;
        float w01 = (vy0 && vx1) ? wy0 * wx1 * ms : 0.f;
        float w10 = (vy1 && vx0) ? wy1 * wx0 * ms : 0.f;
        float w11 = (vy1 && vx1) ? wy1 * wx1 * ms : 0.f;
        int i00 = cy0 * Ww + cx0, i01 = cy0 * Ww + cx1;
        int i10 = cy1 * Ww + cx0, i11 = cy1 * Ww + cx1;

        const float* xb = x + b * Cc * HW;
        float* vrow = valLds + m * VAL_STRIDE + tap;
        for (int c = 0; c < Cc; ++c) {
            const float* xc = xb + c * HW;
            float v = w00 * xc[i00] + w01 * xc[i01] + w10 * xc[i10] + w11 * xc[i11];
            vrow[c * 9] = v;
        }
    }

    // all of this wave's async weight copies have landed in LDS
    asm volatile("s_wait_asynccnt 0" ::: "memory");
    __syncthreads();

    // ---- WMMA: D = A(16x576) x B(576x16) accumulated over 144 k-steps ----
    const int wv   = tid >> 5;
    const int lane = tid & 31;
    const int msub = wv >> 2;       // 0..1
    const int nw   = wv & 3;        // 0..3

    v8f acc = {};
    const float* arow = valLds + (msub * 16 + (lane & 15)) * VAL_STRIDE + 2 * (lane >> 4);
    const v2f*   brow = wLds + nw * NSTEP * 32 + lane;
    for (int s = 0; s < NSTEP; ++s) {
        v2f a  = *(const v2f*)(arow + 4 * s);       // A[m][k0], A[m][k0+1]
        v2f bb = brow[s * 32];                      // B[k0][n], B[k0+1][n]
        acc = __builtin_amdgcn_wmma_f32_16x16x4_f32(
            /*neg_a=*/false, a, /*neg_b=*/false, bb,
            /*c_mod=*/(short)0, acc, /*reuse_a=*/false, /*reuse_b=*/false);
    }

    // ---- store pre-BN result in [pixel][channel] layout ----
    const int n     = nw * 16 + (lane & 15);
    const int mbase = gm0 + msub * 16 + 8 * (lane >> 4);   // C/D: vgpr r -> row r + 8*(lane/16)
#pragma unroll
    for (int r = 0; r < 8; ++r)
        pre[(mbase + r) * Oo + n] = acc[r];
}

// ---------------------------------------------------------------------------
// Kernel 3: per-channel batch stats -> fused scale/shift
// ---------------------------------------------------------------------------
__global__ __launch_bounds__(256)
void dcn_bn_stats_kernel(const float* __restrict__ pre,
                         const float* __restrict__ gamma, const float* __restrict__ beta,
                         float* __restrict__ scale, float* __restrict__ shift) {
    __shared__ float ssum[256], ssq[256];
    const int o = blockIdx.x, tid = threadIdx.x;
    float s = 0.f, s2 = 0.f;
    for (int i = tid; i < 2 * HW; i += 256) {
        float v = pre[i * Oo + o];
        s += v; s2 += v * v;
    }
    ssum[tid] = s; ssq[tid] = s2;
    __syncthreads();
    for (int off = 128; off; off >>= 1) {
        if (tid < off) { ssum[tid] += ssum[tid + off]; ssq[tid] += ssq[tid + off]; }
        __syncthreads();
    }
    if (tid == 0) {
        const float invN = 1.f / (2.f * (float)HW);
        float mean = ssum[0] * invN;
        float var  = ssq[0] * invN - mean * mean;
        float sc   = gamma[o] * rsqrtf(var + 1e-5f);
        scale[o] = sc;
        shift[o] = beta[o] - mean * sc;
    }
}

// ---------------------------------------------------------------------------
// Kernel 4: affine + ReLU, transpose [pixel][o] -> NCHW output
// ---------------------------------------------------------------------------
__global__ __launch_bounds__(256)
void dcn_bn_apply_kernel(const float* __restrict__ pre,
                         const float* __restrict__ scale, const float* __restrict__ shift,
                         float* __restrict__ out) {
    int idx = blockIdx.x * 256 + threadIdx.x;   // < 2*64*HW = 2^21
    int b   = idx >> 20;
    int rem = idx & ((1 << 20) - 1);
    int o   = rem >> 14;
    int hw  = rem & (HW - 1);
    int gm  = (b << 14) + hw;
    float v = pre[gm * Oo + o] * scale[o] + shift[o];
    out[idx] = fmaxf(v, 0.f);
}

// ---------------------------------------------------------------------------
extern "C" void kernel_launch(void* const* d_in, const int* in_sizes, int n_in,
                              void* d_out, int out_size, void* d_ws, size_t ws_size,
                              hipStream_t stream) {
    const float* x      = (const float*)d_in[0];
    const float* conv_w = (const float*)d_in[1];
    const float* off_w  = (const float*)d_in[2];
    const float* off_b  = (const float*)d_in[3];
    const float* mask_w = (const float*)d_in[4];
    const float* mask_b = (const float*)d_in[5];
    const float* gamma  = (const float*)d_in[6];
    const float* beta   = (const float*)d_in[7];

    float* ws    = (float*)d_ws;
    float* offo  = ws;                       // 2*18*HW  = 589824 floats
    float* masko = offo + 2 * 18 * HW;       // 2*9*HW   = 294912 floats
    float* pre   = masko + 2 * 9 * HW;       // 2*64*HW  = 2097152 floats ([pixel][o])
    float* scale = pre + 2 * Oo * HW;        // 64
    float* shift = scale + Oo;               // 64
    float* out   = (float*)d_out;            // [B,64,H,W] f32

    dcn_offset_mask_kernel<<<(2 * HW) / 256, 256, 0, stream>>>(
        x, off_w, off_b, mask_w, mask_b, offo, masko);

    const size_t lds_bytes = (size_t)(VAL_FLOATS + 2 * WFRAG_COUNT) * sizeof(float); // 221,696 B
    dcn_wmma_kernel<<<(2 * HW) / MT, 256, lds_bytes, stream>>>(
        x, conv_w, offo, masko, pre);

    dcn_bn_stats_kernel<<<Oo, 256, 0, stream>>>(pre, gamma, beta, scale, shift);

    dcn_bn_apply_kernel<<<(2 * Oo * HW) / 256, 256, 0, stream>>>(pre, scale, shift, out);
}